// CosineAttentionClassifier_20538533609518
// MI455X (gfx1250) — compile-verified
//
#include <hip/hip_runtime.h>
#include <cstdint>

// Problem constants (from reference setup_inputs): s[N,B,D], target[B,D] -> out[B,N]
#define N_ 32
#define B_ 2048
#define D_ 1024
#define EPSF 1e-10f

typedef float v2f __attribute__((ext_vector_type(2)));
typedef float v8f __attribute__((ext_vector_type(8)));

// One 256-thread workgroup (8 wave32) per batch element b.
// Wave w handles n = 4w .. 4w+3 as two n-pairs. Lane l owns d-slices
// {j*128 + l*4 .. +3}, j=0..7 (coalesced 512B per wave-load).
__global__ __launch_bounds__(256)
void cosine_sim_bn_kernel(const float* __restrict__ s,
                          const float* __restrict__ target,
                          float* __restrict__ out)
{
    const int b    = blockIdx.x;
    const int tid  = threadIdx.x;
    const int wave = tid >> 5;
    const int lane = tid & 31;

    __shared__ float tgt[D_];   // 4 KB: target[b, :]

    // ---------------------------------------------------------------
    // Stage 1: async DMA target[b,:] from global into LDS (ASYNCcnt
    // path). Each of the 256 threads moves one b128.
    // ---------------------------------------------------------------
    {
        unsigned lds_off = (unsigned)(reinterpret_cast<uintptr_t>(&tgt[0]))
                         + (unsigned)(tid * 16);
        unsigned voff    = (unsigned)(tid * 16);
        unsigned long long base =
            (unsigned long long)(uintptr_t)(target + (size_t)b * D_);
        asm volatile("global_load_async_to_lds_b128 %0, %1, %2"
                     :
                     : "v"(lds_off), "v"(voff), "s"(base)
                     : "memory");
        asm volatile("s_wait_asynccnt 0" ::: "memory");
    }
    __syncthreads();

    // ---------------------------------------------------------------
    // Stage 2: per-wave register cache of the target slice + its
    // self-norm (shared across this wave's 4 n-values).
    // ---------------------------------------------------------------
    float4 t4[8];
    float  tt = 0.0f;
#pragma unroll
    for (int j = 0; j < 8; ++j) {
        t4[j] = *reinterpret_cast<const float4*>(&tgt[j * 128 + lane * 4]);
        tt = fmaf(t4[j].x, t4[j].x, tt);
        tt = fmaf(t4[j].y, t4[j].y, tt);
        tt = fmaf(t4[j].z, t4[j].z, tt);
        tt = fmaf(t4[j].w, t4[j].w, tt);
    }
#pragma unroll
    for (int off = 16; off > 0; off >>= 1)
        tt += __shfl_xor(tt, off, 32);
    const float rt = rsqrtf(fmaxf(tt, EPSF));   // rsqrt(clip(|t|^2, EPS))

    // WMMA selector B-matrix: column 0 = (1,0,1,0)^T, column 1 = (0,1,0,1)^T.
    // B 4x16 f32 layout: VGPR0 = rows {0,2}, VGPR1 = rows {1,3} across lane
    // halves, column = lane&15.
    v2f Bsel;
    Bsel.x = ((lane & 15) == 0) ? 1.0f : 0.0f;
    Bsel.y = ((lane & 15) == 1) ? 1.0f : 0.0f;
    const v8f Cz = {0.f, 0.f, 0.f, 0.f, 0.f, 0.f, 0.f, 0.f};

    // ---------------------------------------------------------------
    // Stage 3: stream s[n,b,:] once per n, fusing dot(s,t) and |s|^2,
    // two n's at a time; reduce accumulator pairs with one
    // v_wmma_f32_16x16x4_f32 each (exact f32 32-lane reduction).
    // ---------------------------------------------------------------
#pragma unroll
    for (int p = 0; p < 2; ++p) {
        const int n0 = (wave << 2) + (p << 1);
        const float* __restrict__ sp0 = s + ((size_t)n0 * B_ + b) * (size_t)D_;
        const float* __restrict__ sp1 = sp0 + (size_t)B_ * (size_t)D_;

        // Warm this wave's next n-pair (global_prefetch_b8): one 128B
        // line per lane covers each 4 KB row.
        if (p == 0) {
            __builtin_prefetch(sp0 + 2 * (size_t)B_ * D_ + (size_t)lane * 32, 0, 0);
            __builtin_prefetch(sp1 + 2 * (size_t)B_ * D_ + (size_t)lane * 32, 0, 0);
        }

        float dv0 = 0.f, sv0 = 0.f;   // n0: dot, |s|^2 partials
        float dv1 = 0.f, sv1 = 0.f;   // n1: dot, |s|^2 partials
#pragma unroll
        for (int j = 0; j < 8; ++j) {
            const float4 v0 =
                *reinterpret_cast<const float4*>(&sp0[j * 128 + lane * 4]);
            const float4 v1 =
                *reinterpret_cast<const float4*>(&sp1[j * 128 + lane * 4]);
            dv0 = fmaf(v0.x, t4[j].x, dv0); dv0 = fmaf(v0.y, t4[j].y, dv0);
            dv0 = fmaf(v0.z, t4[j].z, dv0); dv0 = fmaf(v0.w, t4[j].w, dv0);
            sv0 = fmaf(v0.x, v0.x, sv0);    sv0 = fmaf(v0.y, v0.y, sv0);
            sv0 = fmaf(v0.z, v0.z, sv0);    sv0 = fmaf(v0.w, v0.w, sv0);
            dv1 = fmaf(v1.x, t4[j].x, dv1); dv1 = fmaf(v1.y, t4[j].y, dv1);
            dv1 = fmaf(v1.z, t4[j].z, dv1); dv1 = fmaf(v1.w, t4[j].w, dv1);
            sv1 = fmaf(v1.x, v1.x, sv1);    sv1 = fmaf(v1.y, v1.y, sv1);
            sv1 = fmaf(v1.z, v1.z, sv1);    sv1 = fmaf(v1.w, v1.w, sv1);
        }

        // A matrix (16x4 f32): VGPR0 (=.x) holds k in {0,2}, VGPR1 (=.y)
        // holds k in {1,3} across lane halves -> with the selector B,
        // column 0 of D reduces {dv0,sv0}, column 1 reduces {dv1,sv1}.
        v2f Ad; Ad.x = dv0; Ad.y = dv1;
        v2f As; As.x = sv0; As.y = sv1;

        v8f Dd = __builtin_amdgcn_wmma_f32_16x16x4_f32(
            false, Ad, false, Bsel, (short)0, Cz, false, false);
        v8f Ds = __builtin_amdgcn_wmma_f32_16x16x4_f32(
            false, As, false, Bsel, (short)0, Cz, false, false);

        // Lane c holds D[rows 0..7, col c]; lane c+16 holds rows 8..15.
        float rd = Dd[0] + Dd[1] + Dd[2] + Dd[3] + Dd[4] + Dd[5] + Dd[6] + Dd[7];
        float rs = Ds[0] + Ds[1] + Ds[2] + Ds[3] + Ds[4] + Ds[5] + Ds[6] + Ds[7];
        rd += __shfl_xor(rd, 16, 32);
        rs += __shfl_xor(rs, 16, 32);

        // lane 0 now holds (dot,|s|^2) totals for n0; lane 1 for n0+1.
        if (lane < 2) {
            out[(size_t)b * N_ + n0 + lane] =
                rd * rt * rsqrtf(fmaxf(rs, EPSF));
        }
    }
}

extern "C" void kernel_launch(void* const* d_in, const int* in_sizes, int n_in,
                              void* d_out, int out_size, void* d_ws, size_t ws_size,
                              hipStream_t stream) {
    (void)in_sizes; (void)n_in; (void)d_ws; (void)ws_size; (void)out_size;
    const float* s      = (const float*)d_in[0];   // [N, B, D] f32
    const float* target = (const float*)d_in[1];   // [B, D]    f32
    // d_in[2] = s_label (int64) — unused by the reference computation.
    float* out = (float*)d_out;                    // [B, N]    f32

    cosine_sim_bn_kernel<<<B_, 256, 0, stream>>>(s, target, out);
}